// SVFLayer_22488448762639
// MI455X (gfx1250) — compile-verified
//
#include <hip/hip_runtime.h>
#include <stdint.h>

// ---------------------------------------------------------------------------
// SVF time-varying 2-state filter, S=64000 steps, B=64 channels.
// Memory-bound (~165 MB moved => ~8us @ 23.3 TB/s). Parallelized as a
// chunked associative scan over affine maps (M,v) per channel:
//   pass 1: per-chunk compose of 64 per-step maps (TDM-staged inputs in LDS)
//   pass 2: Hillis-Steele scan of 1000 chunk maps per channel (in LDS)
//   pass 3: replay each chunk from its exact start state, emit mixed output
//           through an LDS transpose tile for coalesced [B,S] stores.
// CDNA5-specific paths: TENSOR_LOAD_TO_LDS (TDM) + s_wait_tensorcnt.
// ---------------------------------------------------------------------------

#define S_TOTAL 64000
#define B_CH    64
#define L_CHUNK 64
#define C_CHUNKS (S_TOTAL / L_CHUNK)   // 1000
#define NP (B_CH * C_CHUNKS)           // 64000 per-ws-array elements

typedef unsigned int u32x4 __attribute__((ext_vector_type(4)));
typedef int          i32x8 __attribute__((ext_vector_type(8)));
typedef int          i32x4 __attribute__((ext_vector_type(4)));

// 1D contiguous TDM load: n_elems f32 from gptr -> LDS byte offset lds_off.
// D# group0: count=1 | lds_addr | global_addr(57b) | type=2.
// D# group1: data_size=2 (4B); tensor_dim0 = tile_dim0 = n_elems (never OOB);
//            tile_dim1/2 = 0 (1-D); stride0 = n_elems; rest zero.
// This toolchain's builtin takes 6 args: (g0, g1, g2, g3, extra_v8, cpol);
// groups 2/3 and the extra group are unused for a 1-D tile -> zero-filled.
__device__ __forceinline__ void tdm_load_f32_1d(uint32_t lds_off,
                                                const float* gptr,
                                                uint32_t n_elems) {
  uint64_t ga = (uint64_t)(uintptr_t)gptr;
  u32x4 g0;
  g0[0] = 1u;                                             // count=1, user D#
  g0[1] = lds_off;                                        // LDS byte address
  g0[2] = (uint32_t)ga;                                   // global addr lo
  g0[3] = ((uint32_t)(ga >> 32) & 0x01FFFFFFu) | (2u << 30); // addr hi | type=2
  i32x8 g1;
  g1[0] = (int)(2u << 16);                                // data_size = 4 bytes
  g1[1] = (int)((n_elems & 0xFFFFu) << 16);               // tensor_dim0[15:0]
  g1[2] = (int)(((n_elems >> 16) & 0xFFFFu) | (1u << 16));// td0[31:16], td1=1
  g1[3] = (int)((n_elems & 0xFFFFu) << 16);               // tile_dim0
  g1[4] = 0;                                              // tile_dim1/2 = 0
  g1[5] = (int)n_elems;                                   // dim0 stride lo
  g1[6] = 0;
  g1[7] = 0;
  i32x4 z4; z4[0] = z4[1] = z4[2] = z4[3] = 0;            // groups 2/3 unused
  i32x8 z8;
  z8[0] = z8[1] = z8[2] = z8[3] = 0;
  z8[4] = z8[5] = z8[6] = z8[7] = 0;
  __builtin_amdgcn_tensor_load_to_lds(g0, g1, z4, z4, z8, 0);
}

// ---- Pass 1: per-(chunk, channel) composition of 64 step maps -------------
__global__ __launch_bounds__(B_CH) void svf_chunk_reduce(
    const float* __restrict__ audio, const float* __restrict__ g,
    const float* __restrict__ twoR, float* __restrict__ ws) {
  __shared__ float sm[3 * L_CHUNK * B_CH];                // g | twoR | audio
  const int c  = blockIdx.x;
  const int ch = threadIdx.x;
  const uint32_t base = (uint32_t)c * (L_CHUNK * B_CH);

  if (threadIdx.x == 0) {
    tdm_load_f32_1d(0u,                       g     + base, L_CHUNK * B_CH);
    tdm_load_f32_1d(L_CHUNK * B_CH * 4u,      twoR  + base, L_CHUNK * B_CH);
    tdm_load_f32_1d(2u * L_CHUNK * B_CH * 4u, audio + base, L_CHUNK * B_CH);
    __builtin_amdgcn_s_wait_tensorcnt(0);
  }
  __syncthreads();

  const float* sg = sm;
  const float* sr = sm + L_CHUNK * B_CH;
  const float* sx = sm + 2 * L_CHUNK * B_CH;

  // state map: s_out = M s_in + v  (per-step A = 2H - I, bb = 2*gHBx)
  float m00 = 1.f, m01 = 0.f, m10 = 0.f, m11 = 1.f, v0 = 0.f, v1 = 0.f;
#pragma unroll 4
  for (int t = 0; t < L_CHUNK; ++t) {
    float gv = sg[t * B_CH + ch];
    float rv = sr[t * B_CH + ch];
    float xv = sx[t * B_CH + ch];
    float T   = 1.0f / (1.0f + gv * (gv + rv));
    float gT  = gv * T;
    float a00 = 2.f * T - 1.f;
    float a01 = -2.f * gT;
    float a10 = 2.f * gT;
    float a11 = 2.f * T * (rv * gv + 1.f) - 1.f;
    float bb0 = 2.f * gT * xv;
    float bb1 = gv * bb0;
    float n00 = a00 * m00 + a01 * m10;
    float n01 = a00 * m01 + a01 * m11;
    float n10 = a10 * m00 + a11 * m10;
    float n11 = a10 * m01 + a11 * m11;
    float w0  = a00 * v0 + a01 * v1 + bb0;
    float w1  = a10 * v0 + a11 * v1 + bb1;
    m00 = n00; m01 = n01; m10 = n10; m11 = n11; v0 = w0; v1 = w1;
  }
  const int idx = ch * C_CHUNKS + c;
  ws[0 * NP + idx] = m00;
  ws[1 * NP + idx] = m01;
  ws[2 * NP + idx] = m10;
  ws[3 * NP + idx] = m11;
  ws[4 * NP + idx] = v0;
  ws[5 * NP + idx] = v1;
}

// ---- Pass 2: per-channel Hillis-Steele scan over 1000 chunk maps ----------
__global__ __launch_bounds__(1024) void svf_chunk_scan(float* __restrict__ ws) {
  const int ch = blockIdx.x;
  const int j  = threadIdx.x;
  __shared__ float sm00[1024], sm01[1024], sm10[1024], sm11[1024];
  __shared__ float sv0[1024], sv1[1024];

  float m00 = 1.f, m01 = 0.f, m10 = 0.f, m11 = 1.f, v0 = 0.f, v1 = 0.f;
  if (j < C_CHUNKS) {
    const int idx = ch * C_CHUNKS + j;
    m00 = ws[0 * NP + idx]; m01 = ws[1 * NP + idx];
    m10 = ws[2 * NP + idx]; m11 = ws[3 * NP + idx];
    v0  = ws[4 * NP + idx]; v1  = ws[5 * NP + idx];
  }
  sm00[j] = m00; sm01[j] = m01; sm10[j] = m10; sm11[j] = m11;
  sv0[j] = v0;  sv1[j] = v1;
  __syncthreads();

  for (int d = 1; d < 1024; d <<= 1) {
    float p00 = 1.f, p01 = 0.f, p10 = 0.f, p11 = 1.f, q0 = 0.f, q1 = 0.f;
    if (j >= d) {
      p00 = sm00[j - d]; p01 = sm01[j - d];
      p10 = sm10[j - d]; p11 = sm11[j - d];
      q0  = sv0[j - d];  q1  = sv1[j - d];
    }
    __syncthreads();
    if (j >= d) {
      // combined = cur ∘ prev : M = Mc*Mp ; v = Mc*vp + vc
      float n00 = m00 * p00 + m01 * p10;
      float n01 = m00 * p01 + m01 * p11;
      float n10 = m10 * p00 + m11 * p10;
      float n11 = m10 * p01 + m11 * p11;
      float w0  = m00 * q0 + m01 * q1 + v0;
      float w1  = m10 * q0 + m11 * q1 + v1;
      m00 = n00; m01 = n01; m10 = n10; m11 = n11; v0 = w0; v1 = w1;
      sm00[j] = m00; sm01[j] = m01; sm10[j] = m10; sm11[j] = m11;
      sv0[j] = v0;  sv1[j] = v1;
    }
    __syncthreads();
  }

  if (j < C_CHUNKS) {
    float s0 = 1.f, s1 = 1.f;                 // reference state0 = ones
    if (j > 0) {                              // exclusive prefix applied to s0
      float P00 = sm00[j - 1], P01 = sm01[j - 1];
      float P10 = sm10[j - 1], P11 = sm11[j - 1];
      s0 = P00 + P01 + sv0[j - 1];
      s1 = P10 + P11 + sv1[j - 1];
    }
    const int idx = ch * C_CHUNKS + j;
    ws[6 * NP + idx] = s0;
    ws[7 * NP + idx] = s1;
  }
}

// ---- Pass 3: replay each chunk from exact start state, emit output --------
__global__ __launch_bounds__(B_CH) void svf_emit(
    const float* __restrict__ audio, const float* __restrict__ g,
    const float* __restrict__ twoR, const float* __restrict__ mix,
    const float* __restrict__ ws, float* __restrict__ out) {
  // LDS: g[4096] | twoR[4096] | audio[4096] | mix[12288] | y[64*65]
  __shared__ float sm[3 * L_CHUNK * B_CH + 3 * L_CHUNK * B_CH + B_CH * 65];
  const int c  = blockIdx.x;
  const int ch = threadIdx.x;
  const uint32_t base = (uint32_t)c * (L_CHUNK * B_CH);

  if (threadIdx.x == 0) {
    tdm_load_f32_1d(0u,     g     + base,     L_CHUNK * B_CH);
    tdm_load_f32_1d(16384u, twoR  + base,     L_CHUNK * B_CH);
    tdm_load_f32_1d(32768u, audio + base,     L_CHUNK * B_CH);
    tdm_load_f32_1d(49152u, mix   + base * 3, 3 * L_CHUNK * B_CH);
    __builtin_amdgcn_s_wait_tensorcnt(0);
  }
  __syncthreads();

  const float* sg  = sm;
  const float* sr  = sm + 4096;
  const float* sx  = sm + 8192;
  const float* smx = sm + 12288;
  float*       sy  = sm + 12288 + 12288;     // [t][ch], row stride 65

  float s0 = ws[6 * NP + ch * C_CHUNKS + c];
  float s1 = ws[7 * NP + ch * C_CHUNKS + c];

#pragma unroll 4
  for (int t = 0; t < L_CHUNK; ++t) {
    float gv = sg[t * B_CH + ch];
    float rv = sr[t * B_CH + ch];
    float xv = sx[t * B_CH + ch];
    float T  = 1.0f / (1.0f + gv * (gv + rv));
    float gT = gv * T;
    float b0 = gT * xv;
    float b1 = gv * gT * xv;
    float Y0 = b0 + T * s0 - gT * s1;              // H = T*[[1,-g],[g,2Rg+1]]
    float Y1 = b1 + gT * s0 + T * (rv * gv + 1.f) * s1;
    float mx0 = smx[(t * B_CH + ch) * 3 + 0];
    float mx1 = smx[(t * B_CH + ch) * 3 + 1];
    float mx2 = smx[(t * B_CH + ch) * 3 + 2];
    float yh = xv - rv * Y0 - Y1;
    float y  = rv * mx0 * Y0 + mx1 * Y1 + mx2 * yh;
    sy[t * 65 + ch] = y;
    s0 = 2.f * Y0 - s0;
    s1 = 2.f * Y1 - s1;
  }
  __syncthreads();

  // Transposed, coalesced store: out[row*S + t0 + tid]
  const int t0 = c * L_CHUNK;
#pragma unroll 4
  for (int row = 0; row < B_CH; ++row) {
    out[row * S_TOTAL + t0 + ch] = sy[ch * 65 + row];
  }
}

extern "C" void kernel_launch(void* const* d_in, const int* in_sizes, int n_in,
                              void* d_out, int out_size, void* d_ws, size_t ws_size,
                              hipStream_t stream) {
  const float* audio = (const float*)d_in[0];   // [S, B]
  const float* g     = (const float*)d_in[1];   // [S, B, 1]
  const float* twoR  = (const float*)d_in[2];   // [S, B, 1]
  const float* mix   = (const float*)d_in[3];   // [S, B, 3]
  float* out = (float*)d_out;                   // [B, S]
  float* ws  = (float*)d_ws;                    // 8*NP floats (~2 MB)

  svf_chunk_reduce<<<C_CHUNKS, B_CH, 0, stream>>>(audio, g, twoR, ws);
  svf_chunk_scan<<<B_CH, 1024, 0, stream>>>(ws);
  svf_emit<<<C_CHUNKS, B_CH, 0, stream>>>(audio, g, twoR, mix, ws, out);
}